// HierarchicalDiscoMixerBlock_26869315403892
// MI455X (gfx1250) — compile-verified
//
#include <hip/hip_runtime.h>
#include <math.h>

// ---------------- CDNA5 WMMA types ----------------
typedef float v2f __attribute__((ext_vector_type(2)));
typedef float v8f __attribute__((ext_vector_type(8)));

// ============================================================
// Generic strided, batched FP32 WMMA GEMM:
//   C[z,m,n] (accum?+=:=) alpha * sum_k A[z,m,k]*B[z,k,n]
// A element (m,k) at A + z*baz + m*sam + k*sak   (likewise B, C)
// Block = 4 waves; each wave owns one 16x16 tile along N, all waves
// share the 16-row A tile which is staged through LDS (k-major) in
// 32-deep K chunks.  Main path carries no masks (zero-fill / clamped
// columns handle edges), so EXEC is all-1s across every WMMA.
// ============================================================
__global__ __launch_bounds__(128)
void wmma_gemm_f32(const float* __restrict__ A, const float* __restrict__ B,
                   float* __restrict__ C,
                   int M, int N, int K,
                   long sam, long sak, long sbk, long sbn, long scm, long scn,
                   long baz, long bbz, long bcz,
                   float alpha, int accum) {
  __shared__ float As[32][16];                // 2KB, k-major: As[k][m]
  const int tid  = threadIdx.x;
  const int lane = tid & 31;
  const int wv   = tid >> 5;                  // wave id -> N tile
  const int mi   = lane & 15;
  const int kh   = lane >> 4;                 // 0/1: K half within step-of-4
  const long zb  = blockIdx.z;
  const float* Ab = A + zb * baz;
  const float* Bb = B + zb * bbz;
  float* Cb       = C + zb * bcz;
  const int m0 = blockIdx.y << 4;
  const int n0 = (blockIdx.x * 4 + wv) << 4;

  // A-fill coordinates: thread loads 4 elements (k = fk+8j, row fm)
  const int fm = tid & 15;
  const int fk = tid >> 4;                    // 0..7
  int gm = m0 + fm; gm = (gm < M) ? gm : (M - 1);   // clamped rows: their
  const float* Afill = Ab + (long)gm * sam;         // outputs are never stored

  const int bn  = n0 + mi;
  const int bnc = (bn < N) ? bn : (N - 1);          // clamped cols: ditto
  const float* Bcol = Bb + (long)bnc * sbn + (long)(kh << 1) * sbk;

  v8f acc = {0.f, 0.f, 0.f, 0.f, 0.f, 0.f, 0.f, 0.f};

  for (int k0 = 0; k0 < K; k0 += 32) {
    // ---- stage A chunk (16 rows x 32 k) into LDS, zero-padded in k ----
    #pragma unroll
    for (int j = 0; j < 4; ++j) {
      const int kk = fk + (j << 3);
      const int gk = k0 + kk;
      As[kk][fm] = (gk < K) ? Afill[(long)gk * sak] : 0.0f;
    }
    __syncthreads();

    const float* bp = Bcol + (long)k0 * sbk;
    __builtin_prefetch(Afill + (long)(k0 + 32) * sak, 0, 3);

    if (k0 + 32 <= K) {                       // fast path: no masks at all
      #pragma unroll
      for (int j = 0; j < 8; ++j) {
        v2f a, b;
        a.x = As[(j << 2) + (kh << 1)][mi];
        a.y = As[(j << 2) + (kh << 1) + 1][mi];
        b.x = bp[(long)(j << 2) * sbk];
        b.y = bp[(long)((j << 2) + 1) * sbk];
        acc = __builtin_amdgcn_wmma_f32_16x16x4_f32(
            false, a, false, b, (short)0, acc, false, false);
      }
    } else {                                  // K tail: mask B, A already 0
      #pragma unroll
      for (int j = 0; j < 8; ++j) {
        const int kk = k0 + (j << 2) + (kh << 1);
        const float mk0 = (kk < K) ? 1.0f : 0.0f;
        const float mk1 = (kk + 1 < K) ? 1.0f : 0.0f;
        const long osafe = (long)(K - 1 - k0 - (kh << 1)) * sbk; // in-range
        const long o0 = (kk < K) ? (long)(j << 2) * sbk : osafe;
        const long o1 = (kk + 1 < K) ? (long)((j << 2) + 1) * sbk : osafe;
        v2f a, b;
        a.x = As[(j << 2) + (kh << 1)][mi];
        a.y = As[(j << 2) + (kh << 1) + 1][mi];
        b.x = bp[o0] * mk0;
        b.y = bp[o1] * mk1;
        acc = __builtin_amdgcn_wmma_f32_16x16x4_f32(
            false, a, false, b, (short)0, acc, false, false);
      }
    }
    __syncthreads();
  }

  // D layout: VGPR r -> row = r + 8*kh, col = mi
  #pragma unroll
  for (int r = 0; r < 8; ++r) {
    const int m = m0 + r + (kh << 3);
    const int n = n0 + mi;
    if (m < M && n < N) {
      const long idx = (long)m * scm + (long)n * scn;
      const float v = alpha * acc[r];
      Cb[idx] = accum ? (Cb[idx] + v) : v;
    }
  }
}

// ---------------- DFT matrices (exact arg reduction mod 128) ----------------
__global__ void k_build_dft(float* Af, float* Sf, float* Ci, float* Si) {
  int t = blockIdx.x * blockDim.x + threadIdx.x;
  if (t >= 65 * 128) return;
  int f = t >> 7, p = t & 127;
  int im = (f * p) & 127;                        // e^{-i 2pi f p /128} periodic
  float ang = (float)im * 0.049087385212340526f; // 2*pi/128
  float c = cosf(ang), s = sinf(ang);
  Af[t] = c;                                     // forward cos
  Sf[t] = s;                                     // forward sin
  float wf = (f == 0 || f == 64) ? 1.0f : 2.0f;  // rfft hermitian weights
  Ci[t] = wf * c * (1.0f / 128.0f);              // inverse cos
  Si[t] = -wf * s * (1.0f / 128.0f);             // inverse sin
}

// ---------------- permutes / pointwise ----------------
// u0[(i*64+b)*128+p] = x[(i*128+p)*66 + b],  b<64
// Strided gather routed through the CDNA5 async global->LDS path
// (GLOBAL_LOAD_ASYNC_TO_LDS_B32, tracked by ASYNCcnt), then contiguous store.
__global__ void k_perm_u0(const float* __restrict__ x, float* __restrict__ u) {
  __shared__ float stage[256];
  int t = blockIdx.x * blockDim.x + threadIdx.x;   // grid is exact multiple
  int p = t & 127, b = (t >> 7) & 63, i = t >> 13;
  const float* src = &x[((long)i * 128 + p) * 66 + b];
  unsigned loff = (unsigned)(uintptr_t)&stage[threadIdx.x];  // LDS byte offset
  asm volatile("global_load_async_to_lds_b32 %0, %1, off"
               :: "v"(loff), "v"(src) : "memory");
  asm volatile("s_wait_asynccnt 0x0" ::: "memory");
  u[t] = stage[threadIdx.x];
}

// u[(i*512 + b*8 + c)*128 + p] = cprev[c][b][i][p]   (cprev: [8][64][64][128])
__global__ void k_perm_unext(const float* __restrict__ cp, float* __restrict__ u) {
  int t = blockIdx.x * blockDim.x + threadIdx.x;
  if (t >= 64 * 512 * 128) return;
  int p = t & 127;
  int r = t >> 7;
  int c = r & 7, b = (r >> 3) & 63, i = r >> 9;
  u[t] = cp[(long)c * 524288 + (long)b * 8192 + i * 128 + p];
}

// Y[fc][o][b*64+l] = sum_{c,k} w[o,c,k] * Z[fc][(k*64+l)][(b*cin+c)]
__global__ void k_apply_w(const float* __restrict__ Zr, const float* __restrict__ Zi,
                          const float* __restrict__ w,
                          float* __restrict__ Yr, float* __restrict__ Yi,
                          int cin, int FC, int BC) {
  long t = (long)blockIdx.x * blockDim.x + threadIdx.x;
  long total = (long)FC * 8 * 4096;
  if (t >= total) return;
  int bl = (int)(t & 4095);
  int o  = (int)((t >> 12) & 7);
  int fc = (int)(t >> 15);
  int l = bl & 63, b = bl >> 6;
  long zbase = (long)fc * 448 * BC;
  float sr = 0.f, si = 0.f;
  for (int c = 0; c < cin; ++c)
    for (int k = 0; k < 7; ++k) {
      float wv = w[(o * cin + c) * 7 + k];
      long zi_ = zbase + (long)(k * 64 + l) * BC + b * cin + c;
      sr += wv * Zr[zi_];
      si += wv * Zi[zi_];
    }
  long yi_ = (long)fc * 32768 + (long)o * 4096 + bl;
  Yr[yi_] = sr;
  Yi[yi_] = si;
}

__global__ void k_bias_c(float* __restrict__ c, const float* __restrict__ b) {
  int t = blockIdx.x * blockDim.x + threadIdx.x;
  if (t >= 8 * 524288) return;
  c[t] += b[t >> 19];
}

__device__ __forceinline__ float gelu_exact(float x) {
  return 0.5f * x * (1.0f + erff(x * 0.70710678118654752f));
}

// h[r][m] = gelu(h + hb1n[m]),  h: [131072][32]
__global__ void k_gelu_h(float* __restrict__ h, const float* __restrict__ hb) {
  int t = blockIdx.x * blockDim.x + threadIdx.x;
  if (t >= 131072 * 32) return;
  h[t] = gelu_exact(h[t] + hb[t & 31]);
}

// ho_n[r] = sum_m h[r][m]*hw2n[m] + hb2[n]
__global__ void k_headout(const float* __restrict__ h, const float* __restrict__ hw2n,
                          const float* __restrict__ hb2, int n, float* __restrict__ ho) {
  int t = blockIdx.x * blockDim.x + threadIdx.x;
  if (t >= 131072) return;
  float s = hb2[n];
  const float* hr = h + (long)t * 32;
  #pragma unroll
  for (int m = 0; m < 32; ++m) s += hr[m] * hw2n[m];
  ho[t] = s;
}

// xfull[s][d] = x[s][d] + (d<64 ? ho[n][dd*8192 + s] : 0),  n=d/16, dd=d%16
__global__ void k_xfull(const float* __restrict__ x, const float* __restrict__ ho,
                        float* __restrict__ xf) {
  int t = blockIdx.x * blockDim.x + threadIdx.x;
  if (t >= 8192 * 66) return;
  int d = t % 66, s = t / 66;
  float v = x[t];
  if (d < 64) {
    int n = d >> 4, dd = d & 15;
    v += ho[(long)n * 131072 + (long)dd * 8192 + s];
  }
  xf[t] = v;
}

__global__ void k_gelu_g(float* __restrict__ g, const float* __restrict__ fb0) {
  int t = blockIdx.x * blockDim.x + threadIdx.x;
  if (t >= 8192 * 128) return;
  g[t] = gelu_exact(g[t] + fb0[t & 127]);
}

// out[s][d] = d<64 ? f2[s][d] + fb1[d] + xfull[s][d] : xfull[s][d]
__global__ void k_final(const float* __restrict__ f2, const float* __restrict__ fb1,
                        const float* __restrict__ xf, float* __restrict__ out) {
  int t = blockIdx.x * blockDim.x + threadIdx.x;
  if (t >= 8192 * 66) return;
  int d = t % 66, s = t / 66;
  out[t] = (d < 64) ? (f2[(long)s * 64 + d] + fb1[d] + xf[t]) : xf[t];
}

// ---------------- host side ----------------
static void launch_gemm(hipStream_t st, const float* A, const float* B, float* C,
                        int M, int N, int K,
                        long sam, long sak, long sbk, long sbn, long scm, long scn,
                        long baz, long bbz, long bcz, int batch,
                        float alpha, int accum) {
  dim3 grid((N + 63) / 64, (M + 15) / 16, batch);
  dim3 block(128);
  wmma_gemm_f32<<<grid, block, 0, st>>>(A, B, C, M, N, K, sam, sak, sbk, sbn,
                                        scm, scn, baz, bbz, bcz, alpha, accum);
}
static inline dim3 g1(long n) { return dim3((unsigned)((n + 255) / 256)); }

// workspace offsets (floats); peak ~140MB with tail-phase aliasing
static const long OFF_AF = 0;          // 65*128
static const long OFF_SF = 8320;
static const long OFF_CI = 16640;
static const long OFF_SI = 24960;
static const long OFF_U  = 33280;      // 4194304
static const long OFF_XR = 4227584;    // 2129920
static const long OFF_XI = 6357504;    // 2129920
static const long OFF_PR = 8487424;    // 1863680
static const long OFF_PI = 10351104;   // 1863680
static const long OFF_ZR = 12214784;   // 2981888 (13-freq chunk)
static const long OFF_ZI = 15196672;   // 2981888
static const long OFF_YR = 18178560;   // 2129920
static const long OFF_YI = 20308480;   // 2129920
static const long OFF_C0 = 22438400;   // 4194304 each
static const long OFF_C1 = 26632704;
static const long OFF_C2 = 30827008;   // end 35021312 floats (~140MB)
// tail-phase aliases (conv scratch no longer live)
static const long OFF_HB = OFF_U;      // 131072*32
static const long OFF_HO = OFF_XR;     // 4*131072
static const long OFF_XF = OFF_XI;     // 8192*66
static const long OFF_G  = OFF_PR;     // 8192*128
static const long OFF_F2 = OFF_PI;     // 8192*64

extern "C" void kernel_launch(void* const* d_in, const int* in_sizes, int n_in,
                              void* d_out, int out_size, void* d_ws, size_t ws_size,
                              hipStream_t stream) {
  (void)in_sizes; (void)n_in; (void)out_size; (void)ws_size;
  const float* x = (const float*)d_in[0];
  const float* psi[3] = {(const float*)d_in[1], (const float*)d_in[2], (const float*)d_in[3]};
  const float* wl[3]  = {(const float*)d_in[4], (const float*)d_in[6], (const float*)d_in[8]};
  const float* bl[3]  = {(const float*)d_in[5], (const float*)d_in[7], (const float*)d_in[9]};
  const float* hw1 = (const float*)d_in[10];
  const float* hb1 = (const float*)d_in[11];
  const float* hw2 = (const float*)d_in[12];
  const float* hb2 = (const float*)d_in[13];
  const float* fw0 = (const float*)d_in[14];
  const float* fb0 = (const float*)d_in[15];
  const float* fw1 = (const float*)d_in[16];
  const float* fb1 = (const float*)d_in[17];
  float* out = (float*)d_out;
  float* W = (float*)d_ws;

  k_build_dft<<<g1(65 * 128), 256, 0, stream>>>(W + OFF_AF, W + OFF_SF, W + OFF_CI, W + OFF_SI);
  k_perm_u0<<<g1(64 * 64 * 128), 256, 0, stream>>>(x, W + OFF_U);

  const long cOff[3] = {OFF_C0, OFF_C1, OFF_C2};
  const int cins[3] = {1, 8, 8};

  for (int lvl = 0; lvl < 3; ++lvl) {
    const int cin = cins[lvl];
    const int BC = 64 * cin;       // columns: b*cin + c
    const int NX = 64 * BC;        // rows of X per frequency block

    // Ph = conj(rfft(psi)):  Pr[f][(k,l,i)], Pi[f][(k,l,i)]
    launch_gemm(stream, W + OFF_AF, psi[lvl], W + OFF_PR, 65, 28672, 128,
                128, 1, 1, 128, 28672, 1, 0, 0, 0, 1, 1.0f, 0);
    launch_gemm(stream, W + OFF_SF, psi[lvl], W + OFF_PI, 65, 28672, 128,
                128, 1, 1, 128, 28672, 1, 0, 0, 0, 1, 1.0f, 0);

    // Xh = rfft(u): Xr/Xi laid out [f][i][bc]
    launch_gemm(stream, W + OFF_AF, W + OFF_U, W + OFF_XR, 65, NX, 128,
                128, 1, 1, 128, NX, 1, 0, 0, 0, 1, 1.0f, 0);
    launch_gemm(stream, W + OFF_SF, W + OFF_U, W + OFF_XI, 65, NX, 128,
                128, 1, 1, 128, NX, 1, 0, 0, 0, 1, -1.0f, 0);

    // per-frequency Zh = Ph*Xh in 13-freq chunks, then fold w -> Y[f][o][b*64+l]
    for (int f0 = 0; f0 < 65; f0 += 13) {
      const float* Pr = W + OFF_PR + (long)f0 * 28672;
      const float* Pi = W + OFF_PI + (long)f0 * 28672;
      const float* Xr = W + OFF_XR + (long)f0 * NX;
      const float* Xi = W + OFF_XI + (long)f0 * NX;
      // Zr = Pr@Xr - Pi@Xi ; Zi = Pr@Xi + Pi@Xr   (M=448,K=64,N=BC, batch=13)
      launch_gemm(stream, Pr, Xr, W + OFF_ZR, 448, BC, 64, 64, 1, BC, 1, BC, 1,
                  28672, NX, (long)448 * BC, 13, 1.0f, 0);
      launch_gemm(stream, Pi, Xi, W + OFF_ZR, 448, BC, 64, 64, 1, BC, 1, BC, 1,
                  28672, NX, (long)448 * BC, 13, -1.0f, 1);
      launch_gemm(stream, Pr, Xi, W + OFF_ZI, 448, BC, 64, 64, 1, BC, 1, BC, 1,
                  28672, NX, (long)448 * BC, 13, 1.0f, 0);
      launch_gemm(stream, Pi, Xr, W + OFF_ZI, 448, BC, 64, 64, 1, BC, 1, BC, 1,
                  28672, NX, (long)448 * BC, 13, 1.0f, 1);
      k_apply_w<<<g1((long)13 * 8 * 4096), 256, 0, stream>>>(
          W + OFF_ZR, W + OFF_ZI, wl[lvl],
          W + OFF_YR + (long)f0 * 32768, W + OFF_YI + (long)f0 * 32768, cin, 13, BC);
    }

    // irfft: c[o][b][l][p] = sum_f Yr*Ci + Yi*Si   (M=32768,K=65,N=128)
    launch_gemm(stream, W + OFF_YR, W + OFF_CI, W + cOff[lvl], 32768, 128, 65,
                1, 32768, 128, 1, 128, 1, 0, 0, 0, 1, 1.0f, 0);
    launch_gemm(stream, W + OFF_YI, W + OFF_SI, W + cOff[lvl], 32768, 128, 65,
                1, 32768, 128, 1, 128, 1, 0, 0, 0, 1, 1.0f, 1);
    k_bias_c<<<g1(8 * 524288), 256, 0, stream>>>(W + cOff[lvl], bl[lvl]);

    if (lvl < 2)
      k_perm_unext<<<g1(64 * 512 * 128), 256, 0, stream>>>(W + cOff[lvl], W + OFF_U);
  }

  // per-head MLP: h = gelu(feats@hw1 + hb1); ho = h@hw2 + hb2
  for (int n = 0; n < 4; ++n) {
    for (int lvl = 0; lvl < 3; ++lvl) {
      launch_gemm(stream, W + cOff[lvl] + (long)n * 131072, hw1 + n * 768 + lvl * 256,
                  W + OFF_HB, 131072, 32, 8,
                  1, 524288, 32, 1, 32, 1, 0, 0, 0, 1, 1.0f, lvl ? 1 : 0);
    }
    k_gelu_h<<<g1(131072 * 32), 256, 0, stream>>>(W + OFF_HB, hb1 + n * 32);
    k_headout<<<g1(131072), 256, 0, stream>>>(W + OFF_HB, hw2 + n * 32, hb2, n,
                                              W + OFF_HO + (long)n * 131072);
  }

  // residual + concat -> FFN
  k_xfull<<<g1(8192 * 66), 256, 0, stream>>>(x, W + OFF_HO, W + OFF_XF);
  launch_gemm(stream, W + OFF_XF, fw0, W + OFF_G, 8192, 128, 66,
              66, 1, 128, 1, 128, 1, 0, 0, 0, 1, 1.0f, 0);
  k_gelu_g<<<g1(8192 * 128), 256, 0, stream>>>(W + OFF_G, fb0);
  launch_gemm(stream, W + OFF_G, fw1, W + OFF_F2, 8192, 64, 128,
              128, 1, 64, 1, 64, 1, 0, 0, 0, 1, 1.0f, 0);
  k_final<<<g1(8192 * 66), 256, 0, stream>>>(W + OFF_F2, fb1, W + OFF_XF, out);
}